// ImagePaste_27650999451648
// MI455X (gfx1250) — compile-verified
//
#include <hip/hip_runtime.h>

// ImagePaste: out[b,r,c,ch] = (tl0<=r<br0 && tl1<=c<br1) ? color[b][ch] : 255.0f
// CANVAS=72, BATCH=16384. Output ~1.02 GB fp32 -> pure store-bandwidth bound
// (~44us floor at 23.3 TB/s). Per image there are only TWO distinct 864-byte
// row patterns; stage them in LDS once, then stream the image with CDNA5
// async LDS->global 128-bit stores (ASYNCcnt path), non-temporal.

#define CANVAS    72
#define ROWF      (CANVAS * 3)            // 216 floats per row (864 B)
#define IMGF      (CANVAS * CANVAS * 3)   // 15552 floats per image (62208 B)
#define ROWCHUNKS (ROWF / 4)              // 54 x 16B chunks per row
#define CHUNKS    (CANVAS * ROWCHUNKS)    // 3888 x 16B chunks per image
#define NTHREADS  256                     // 8 wave32 per block

__global__ __launch_bounds__(NTHREADS) void ImagePaste_kernel(
    const int*   __restrict__ pos,   // [B,2,2] int32: tl0,tl1,br0,br1
    const float* __restrict__ col,   // [B,3]
    float*       __restrict__ out)   // [B,72,72,3]
{
    // pat[0..215]   : "active row"  (color inside [tl1,br1), 255 outside)
    // pat[216..431] : "background row" (all 255)
    __shared__ float pat[2 * ROWF];

    const int b   = blockIdx.x;
    const int tid = threadIdx.x;

    const int   tl0 = pos[b * 4 + 0];
    const int   tl1 = pos[b * 4 + 1];
    const int   br0 = pos[b * 4 + 2];
    const int   br1 = pos[b * 4 + 3];
    const float c0  = col[b * 3 + 0];
    const float c1  = col[b * 3 + 1];
    const float c2  = col[b * 3 + 2];

    // Build the two row patterns in LDS.
    for (int e = tid; e < 2 * ROWF; e += NTHREADS) {
        float v = 255.0f;
        if (e < ROWF) {
            int c  = e / 3;          // pixel column
            int ch = e - 3 * c;      // channel
            if (c >= tl1 && c < br1) v = (ch == 0) ? c0 : ((ch == 1) ? c1 : c2);
        }
        pat[e] = v;
    }
    __syncthreads();   // LDS writes visible to async-store reads

    // Workgroup-relative LDS byte offsets of the two patterns.
    const unsigned ldsA = (unsigned)(unsigned long long)(&pat[0]);       // active
    const unsigned ldsB = ldsA + (unsigned)(ROWF * 4);                   // background
    float* imgBase = out + (size_t)b * IMGF;

    // Flat chunk index i = r*54 + k. Grid stride of 256 advances (r,k) by
    // (+4,+40) with at most one carry -> no division in the hot loop.
    int r = tid / ROWCHUNKS;
    int k = tid - r * ROWCHUNKS;
    for (int i = tid; i < CHUNKS; i += NTHREADS) {
        bool inRow = (r >= tl0) && (r < br0);
        unsigned lds = (inRow ? ldsA : ldsB) + (unsigned)k * 16u;
        const float* g = imgBase + (size_t)i * 4;
        asm volatile("global_store_async_from_lds_b128 %0, %1, off th:TH_STORE_NT"
                     :: "v"(g), "v"(lds)
                     : "memory");
        r += 4; k += 40;                       // advance by 256 chunks
        if (k >= ROWCHUNKS) { k -= ROWCHUNKS; r += 1; }
    }

    // Drain outstanding async stores (S_ENDPGM also implies wait-idle).
    asm volatile("s_wait_asynccnt 0" ::: "memory");
}

extern "C" void kernel_launch(void* const* d_in, const int* in_sizes, int n_in,
                              void* d_out, int out_size, void* d_ws, size_t ws_size,
                              hipStream_t stream) {
    const int*   pos = (const int*)d_in[0];    // positions [B,2,2] int32
    const float* col = (const float*)d_in[1];  // colors    [B,3]   float32
    float*       out = (float*)d_out;

    const int batch = in_sizes[1] / 3;         // 16384
    ImagePaste_kernel<<<batch, NTHREADS, 0, stream>>>(pos, col, out);
}